// MaskLinear_22222160789550
// MI455X (gfx1250) — compile-verified
//
#include <hip/hip_runtime.h>
#include <hip/hip_bf16.h>

typedef __attribute__((ext_vector_type(2))) float v2f;
typedef __attribute__((ext_vector_type(8))) float v8f;

#define NCHUNK 2048  // n-elements per block along the reduction dim (multiple of 4)

// Seed output with bias so the atomic accumulation starts from the right value
// every launch (d_out is poisoned once and never re-cleared by the harness).
__global__ void masklinear_init_kernel(const float* __restrict__ bias,
                                       float* __restrict__ out, int total) {
    int i = blockIdx.x * blockDim.x + threadIdx.x;
    if (i < total) out[i] = bias[0];
}

__device__ __forceinline__ v8f wmma_f32x4(v2f a, v2f b, v8f c) {
    return __builtin_amdgcn_wmma_f32_16x16x4_f32(
        /*neg_a=*/false, a, /*neg_b=*/false, b,
        /*c_mod=*/(short)0, c, /*reuse_a=*/false, /*reuse_b=*/false);
}

// out[m,d] += sum_n weight[n]*masks[m,n]*x[n,d] over this block's n-chunk.
// Block: 256 threads = 8 waves. Wave w -> m_tile = w&3 (M=64 -> 4 tiles),
// d_tile = blockIdx.y*2 + (w>>2). DCOMP: compile-time D (0 = runtime).
template <int DCOMP>
__global__ __launch_bounds__(256) void masklinear_wmma_kernel(
    const float* __restrict__ x,       // [N, D]
    const float* __restrict__ masks,   // [M, N]
    const float* __restrict__ weight,  // [N]
    float* __restrict__ out,           // [M, D]
    int N, int Drt)
{
    const int D = (DCOMP > 0) ? DCOMP : Drt;

    const int wave = threadIdx.x >> 5;
    const int lane = threadIdx.x & 31;
    const int l16  = lane & 15;
    const int hi   = lane >> 4;   // 0: lanes 0-15 (K=0,1), 1: lanes 16-31 (K=2,3)

    const int mTile = wave & 3;
    const int dTile = (blockIdx.y << 1) + (wave >> 2);
    const int dBase = dTile * 16;

    const int mRow = mTile * 16 + l16;   // masks row this lane loads (A operand)
    const int dCol = dBase + l16;        // x column this lane loads (B operand)

    const long chunkStart = (long)blockIdx.x * NCHUNK;
    long chunkEnd = chunkStart + NCHUNK;
    if (chunkEnd > N) chunkEnd = N;
    const int span      = (int)(chunkEnd - chunkStart);
    const int fullSteps = span >> 2;     // K-steps of 4
    const int rem       = span & 3;

    // Per-lane streaming pointers (all further addressing via constant offsets).
    const float* __restrict__ mp = masks + (size_t)mRow * (size_t)N + chunkStart + 2 * hi;
    const float* __restrict__ wp = weight + chunkStart + 2 * hi;
    const float* __restrict__ xp = x + ((size_t)chunkStart + 2 * hi) * (size_t)D + dCol;

    v8f acc = {0.f, 0.f, 0.f, 0.f, 0.f, 0.f, 0.f, 0.f};

    // ---- Main loop: 4 K-steps (16 n) per iteration. Issue all loads first,
    // then consume, so the scheduler can overlap VMEM latency with WMMA. ----
    const int batches  = fullSteps >> 2;
    const int stepsRem = fullSteps & 3;

    for (int bt = 0; bt < batches; ++bt) {
        v2f a[4], w[4];
        float b0[4], b1[4];
        #pragma unroll
        for (int s = 0; s < 4; ++s) {
            a[s]  = *(const v2f*)(mp + 4 * s);
            w[s]  = *(const v2f*)(wp + 4 * s);
            b0[s] = xp[(size_t)(4 * s) * D];        // row n0+4s+2*hi
            b1[s] = xp[(size_t)(4 * s + 1) * D];    // row n0+4s+2*hi+1
        }
        mp += 16;
        wp += 16;
        xp += (size_t)16 * D;
        #pragma unroll
        for (int s = 0; s < 4; ++s) {
            v2f af; af.x = a[s].x * w[s].x; af.y = a[s].y * w[s].y;
            v2f bf; bf.x = b0[s]; bf.y = b1[s];
            acc = wmma_f32x4(af, bf, acc);
        }
    }

    // ---- Leftover full K-steps (fullSteps % 4; 0 for N=100000). ----
    for (int s = 0; s < stepsRem; ++s) {
        v2f a = *(const v2f*)mp;
        v2f w = *(const v2f*)wp;
        v2f af; af.x = a.x * w.x; af.y = a.y * w.y;
        v2f bf; bf.x = xp[0]; bf.y = xp[(size_t)D];
        mp += 4;
        wp += 4;
        xp += (size_t)4 * D;
        acc = wmma_f32x4(af, bf, acc);
    }

    // ---- Guarded remainder step (span % 4; 0 for N=100000). ----
    if (rem) {
        const long nA = chunkStart + (long)fullSteps * 4 + 2 * hi;
        const bool v0 = nA < chunkEnd;
        const bool v1 = (nA + 1) < chunkEnd;
        v2f af, bf;
        af.x = v0 ? mp[0] * wp[0] : 0.f;
        af.y = v1 ? mp[1] * wp[1] : 0.f;
        bf.x = v0 ? xp[0] : 0.f;
        bf.y = v1 ? xp[(size_t)D] : 0.f;
        acc = wmma_f32x4(af, bf, acc);
    }

    // C/D layout: VGPR r holds (M = mTile*16 + r + 8*hi, col = dCol).
    #pragma unroll
    for (int r = 0; r < 8; ++r) {
        const int m = mTile * 16 + r + 8 * hi;
        atomicAdd(&out[(size_t)m * (size_t)D + dCol], acc[r]);
    }
}

extern "C" void kernel_launch(void* const* d_in, const int* in_sizes, int n_in,
                              void* d_out, int out_size, void* d_ws, size_t ws_size,
                              hipStream_t stream) {
    const float* x      = (const float*)d_in[0];  // [N, D]
    const float* masks  = (const float*)d_in[1];  // [M, N]
    const float* weight = (const float*)d_in[2];  // [N]
    const float* bias   = (const float*)d_in[3];  // [1]
    float* out          = (float*)d_out;          // [M, D]

    const int N = in_sizes[2];
    const int D = in_sizes[0] / N;                // 256
    // M = in_sizes[1] / N;                       // 64 (tiling assumes M == 64)

    // 1) out = bias
    masklinear_init_kernel<<<(out_size + 255) / 256, 256, 0, stream>>>(bias, out, out_size);

    // 2) atomic-accumulate the contraction
    dim3 grid((N + NCHUNK - 1) / NCHUNK, (unsigned)(D / 32));
    if (D == 256) {
        masklinear_wmma_kernel<256><<<grid, 256, 0, stream>>>(x, masks, weight, out, N, D);
    } else {
        masklinear_wmma_kernel<0><<<grid, 256, 0, stream>>>(x, masks, weight, out, N, D);
    }
}